// AddNoise_24713241821627
// MI455X (gfx1250) — compile-verified
//
#include <hip/hip_runtime.h>
#include <hip/hip_bf16.h>

typedef __attribute__((ext_vector_type(2))) float v2f;
typedef __attribute__((ext_vector_type(4))) float f4;
typedef __attribute__((ext_vector_type(8))) float v8f;

#define T_LEN   262144
#define B_ROWS  64
#define SEG     256
#define NSEG    1024        // T_LEN / SEG
#define C1      0.9763f
#define C3      0.4751f
#define INV_SNR_LIN 0.017782794100389228f  // 1 / 10^(17.5/10)
#define EPS     1e-8f

// workspace layout (float offsets); FF_OFF is 64B aligned for float4 access
#define WS_SUMSQ 0
#define WS_AMAX  1
#define FF_OFF   16
#define ST_OFF   (FF_OFF + T_LEN)          // per-segment end states, NSEG*4
#define SIN_OFF  (ST_OFF + NSEG*4)         // per-segment incoming states, NSEG*4
#define H_OFF    (SIN_OFF + NSEG*4)        // homogeneous response rows, SEG*4
#define PART_OFF (H_OFF + SEG*4)           // power partials, 1024

#define POWER_BLOCKS 1024

// ---------------- init ----------------
__global__ void k_init(float* ws) {
    ws[WS_SUMSQ] = 0.f;
    ((unsigned int*)ws)[WS_AMAX] = 0u;
}

// ---------------- signal power: per-block partials (deterministic) ----------------
__global__ __launch_bounds__(256) void k_power(const float* __restrict__ audio,
                                               float* __restrict__ ws) {
    const f4* a4 = (const f4*)audio;
    const int n4 = B_ROWS * T_LEN / 4;
    float s = 0.f;
    for (int i = blockIdx.x * blockDim.x + threadIdx.x; i < n4;
         i += gridDim.x * blockDim.x) {
        f4 v = a4[i];
        s += v.x * v.x + v.y * v.y + v.z * v.z + v.w * v.w;
    }
    #pragma unroll
    for (int off = 16; off; off >>= 1) s += __shfl_xor(s, off);
    __shared__ float ls[8];
    int lane = threadIdx.x & 31, wv = threadIdx.x >> 5;
    if (lane == 0) ls[wv] = s;
    __syncthreads();
    if (threadIdx.x == 0) {
        float t = 0.f;
        for (int i = 0; i < 8; i++) t += ls[i];
        ws[PART_OFF + blockIdx.x] = t;
    }
}

__global__ __launch_bounds__(256) void k_power_reduce(float* __restrict__ ws) {
    __shared__ float ls[256];
    float s = 0.f;
    for (int i = threadIdx.x; i < POWER_BLOCKS; i += 256) s += ws[PART_OFF + i];
    ls[threadIdx.x] = s;
    __syncthreads();
    for (int off = 128; off; off >>= 1) {
        if (threadIdx.x < off) ls[threadIdx.x] += ls[threadIdx.x + off];
        __syncthreads();
    }
    if (threadIdx.x == 0) ws[WS_SUMSQ] = ls[0];
}

// ---------------- FIR: ff[n] = sum_k b[k] * white[n-k] ----------------
__global__ __launch_bounds__(256) void k_fir(const float* __restrict__ white,
                                             float* __restrict__ ws) {
    int n = blockIdx.x * blockDim.x + threadIdx.x;
    if (n >= T_LEN) return;
    const float b[7] = {0.99886f, 0.0555179f, 0.0750759f, 0.153852f,
                        0.3102f, 0.5329f, -0.0528f};
    float acc = 0.f;
    #pragma unroll
    for (int k = 0; k < 7; k++) {
        int m = n - k;
        if (m >= 0) acc = fmaf(b[k], white[m], acc);
    }
    ws[FF_OFF + n] = acc;
}

// ---------------- per-segment local IIR (zero initial state) ----------------
__global__ __launch_bounds__(256) void k_seg(float* __restrict__ ws) {
    int seg = blockIdx.x * blockDim.x + threadIdx.x;
    if (seg >= NSEG) return;
    float* p = ws + FF_OFF + seg * SEG;
    float y1 = 0.f, y2 = 0.f, y3 = 0.f;
    for (int j = 0; j < SEG; j++) {
        float y = fmaf(C1, y1, fmaf(C3, y3, p[j]));
        p[j] = y;
        y3 = y2; y2 = y1; y1 = y;
    }
    float* st = ws + ST_OFF + seg * 4;
    st[0] = y1; st[1] = y2; st[2] = y3; st[3] = 0.f;
}

// ---------------- 3x3 helpers ----------------
struct M3 { float a[9]; };
__device__ __forceinline__ M3 mmul(const M3& X, const M3& Y) {
    M3 Z;
    #pragma unroll
    for (int r = 0; r < 3; r++)
        #pragma unroll
        for (int c = 0; c < 3; c++)
            Z.a[r*3+c] = X.a[r*3+0]*Y.a[0*3+c] + X.a[r*3+1]*Y.a[1*3+c] + X.a[r*3+2]*Y.a[2*3+c];
    return Z;
}
__device__ __forceinline__ void mv3(const M3& X, const float v[3], float o[3]) {
    #pragma unroll
    for (int r = 0; r < 3; r++)
        o[r] = X.a[r*3+0]*v[0] + X.a[r*3+1]*v[1] + X.a[r*3+2]*v[2];
}

// ---------------- 2-level scan over segment states; also builds H ----------------
__global__ void k_combine(float* __restrict__ ws) {
    __shared__ float wbuf[32][4];
    __shared__ float sbuf[32][4];
    int t = threadIdx.x;  // 0..31, single wave

    if (t == 0) {
        // H_j = C1*H_{j-1} + C3*H_{j-3}; H_{-1}=e0, H_{-2}=e1, H_{-3}=e2
        float h1[3] = {1.f, 0.f, 0.f}, h2[3] = {0.f, 1.f, 0.f}, h3[3] = {0.f, 0.f, 1.f};
        for (int j = 0; j < SEG; j++) {
            float h[3];
            #pragma unroll
            for (int c = 0; c < 3; c++) h[c] = fmaf(C1, h1[c], C3 * h3[c]);
            float* H = ws + H_OFF + j * 4;
            H[0] = h[0]; H[1] = h[1]; H[2] = h[2]; H[3] = 0.f;
            #pragma unroll
            for (int c = 0; c < 3; c++) { h3[c] = h2[c]; h2[c] = h1[c]; h1[c] = h[c]; }
        }
    }

    M3 M = {{C1, 0.f, C3, 1.f, 0.f, 0.f, 0.f, 1.f, 0.f}};
    M3 ML = M;
    #pragma unroll
    for (int i = 0; i < 8; i++) ML = mmul(ML, ML);      // M^256
    M3 ML32 = ML;
    #pragma unroll
    for (int i = 0; i < 5; i++) ML32 = mmul(ML32, ML32); // M^(256*32)

    // level 1: Horner-combine 32 segment contributions per thread
    float w[3] = {0.f, 0.f, 0.f};
    for (int k = 0; k < 32; k++) {
        const float* v = ws + ST_OFF + (t * 32 + k) * 4;
        float tmp[3]; mv3(ML, w, tmp);
        w[0] = tmp[0] + v[0]; w[1] = tmp[1] + v[1]; w[2] = tmp[2] + v[2];
    }
    wbuf[t][0] = w[0]; wbuf[t][1] = w[1]; wbuf[t][2] = w[2];
    __syncthreads();

    // level 2: sequential scan over 32 groups
    if (t == 0) {
        float S[3] = {0.f, 0.f, 0.f};
        sbuf[0][0] = 0.f; sbuf[0][1] = 0.f; sbuf[0][2] = 0.f;
        for (int g = 1; g < 32; g++) {
            float tmp[3]; mv3(ML32, S, tmp);
            #pragma unroll
            for (int c = 0; c < 3; c++) { S[c] = tmp[c] + wbuf[g - 1][c]; sbuf[g][c] = S[c]; }
        }
    }
    __syncthreads();

    // re-walk: emit incoming state for every segment
    float s[3] = {sbuf[t][0], sbuf[t][1], sbuf[t][2]};
    for (int k = 0; k < 32; k++) {
        int seg = t * 32 + k;
        float* si = ws + SIN_OFF + seg * 4;
        si[0] = s[0]; si[1] = s[1]; si[2] = s[2]; si[3] = 0.f;
        const float* v = ws + ST_OFF + seg * 4;
        float tmp[3]; mv3(ML, s, tmp);
        s[0] = tmp[0] + v[0]; s[1] = tmp[1] + v[1]; s[2] = tmp[2] + v[2];
    }
}

// ---------------- WMMA correction: pink += H(256x3) * S(3x1024), tiled 16x16x4 ----------------
// Plus abs-max of the corrected pink into ws[WS_AMAX] (uint-ordered atomicMax).
__global__ __launch_bounds__(256) void k_wmma_corr(float* __restrict__ ws) {
    int wave = (blockIdx.x * blockDim.x + threadIdx.x) >> 5;  // tile id, 0..1023
    int lane = threadIdx.x & 31;
    int jt = wave & 15;   // 16 row tiles (j within segment)
    int st = wave >> 4;   // 64 column tiles (segments)
    int half = lane >> 4;
    int ln = lane & 15;

    const float* H   = ws + H_OFF;
    const float* SIN = ws + SIN_OFF;

    // A (16x4, f32): lanes 0-15 -> {K0,K1}, lanes 16-31 -> {K2,K3}; M = ln
    int row = jt * 16 + ln;
    v2f a;
    a.x = half ? H[row * 4 + 2] : H[row * 4 + 0];
    a.y = half ? 0.f            : H[row * 4 + 1];

    // B (4x16, f32): assume C/D-style row split: VGPR v holds rows K=v (lanes 0-15)
    // and K=v+2 (lanes 16-31); N = ln. K=3 is the zero pad.
    int seg = st * 16 + ln;
    v2f b;
    b.x = half ? SIN[seg * 4 + 2] : SIN[seg * 4 + 0];
    b.y = half ? 0.f              : SIN[seg * 4 + 1];

    // C = local-solution tile: D VGPR r holds M = r + 8*half, N = ln -> contiguous j run
    float* base = ws + FF_OFF + seg * SEG + jt * 16 + half * 8;
    v8f c;
    #pragma unroll
    for (int r = 0; r < 8; r++) c[r] = base[r];

    v8f d = __builtin_amdgcn_wmma_f32_16x16x4_f32(false, a, false, b, (short)0, c,
                                                  false, false);

    float m = 0.f;
    #pragma unroll
    for (int r = 0; r < 8; r++) {
        base[r] = d[r];
        float ad = fabsf(d[r]);
        m = ad > m ? ad : m;
    }
    #pragma unroll
    for (int off = 16; off; off >>= 1) {
        float o = __shfl_xor(m, off);
        m = o > m ? o : m;
    }
    if (lane == 0)
        atomicMax((unsigned int*)ws + WS_AMAX, __float_as_uint(m));
}

// ---------------- final broadcast FMA: out = audio + pink * scale ----------------
__global__ __launch_bounds__(256) void k_final(const float* __restrict__ audio,
                                               const float* __restrict__ ws,
                                               float* __restrict__ out) {
    float sumsq = ws[WS_SUMSQ];
    float amax  = __uint_as_float(((const unsigned int*)ws)[WS_AMAX]);
    float npow  = (sumsq * (1.0f / (float)(B_ROWS * (long long)T_LEN))) * INV_SNR_LIN;
    float scale = sqrtf(npow) / (amax + EPS);

    const f4* a4 = (const f4*)audio;
    const f4* p4 = (const f4*)(ws + FF_OFF);
    f4* o4 = (f4*)out;
    const int n4 = B_ROWS * T_LEN / 4;
    for (int i = blockIdx.x * blockDim.x + threadIdx.x; i < n4;
         i += gridDim.x * blockDim.x) {
        f4 av = a4[i];
        f4 pv = p4[i & (T_LEN / 4 - 1)];
        f4 ov;
        ov.x = fmaf(pv.x, scale, av.x);
        ov.y = fmaf(pv.y, scale, av.y);
        ov.z = fmaf(pv.z, scale, av.z);
        ov.w = fmaf(pv.w, scale, av.w);
        o4[i] = ov;
    }
}

extern "C" void kernel_launch(void* const* d_in, const int* in_sizes, int n_in,
                              void* d_out, int out_size, void* d_ws, size_t ws_size,
                              hipStream_t stream) {
    const float* audio = (const float*)d_in[0];
    const float* white = (const float*)d_in[1];
    float* out = (float*)d_out;
    float* ws  = (float*)d_ws;

    k_init<<<1, 1, 0, stream>>>(ws);
    k_power<<<POWER_BLOCKS, 256, 0, stream>>>(audio, ws);
    k_power_reduce<<<1, 256, 0, stream>>>(ws);
    k_fir<<<T_LEN / 256, 256, 0, stream>>>(white, ws);
    k_seg<<<NSEG / 256, 256, 0, stream>>>(ws);
    k_combine<<<1, 32, 0, stream>>>(ws);
    k_wmma_corr<<<(16 * 64 * 32) / 256, 256, 0, stream>>>(ws);  // 1024 waves = 1024 tiles
    k_final<<<2048, 256, 0, stream>>>(audio, ws, out);
}